// BufCache_4810363372411
// MI455X (gfx1250) — compile-verified
//
#include <hip/hip_runtime.h>
#include <cstdint>

// Sliding-window KV cache append for MI455X (gfx1250).
// ~537 MB HBM traffic, 0 FLOPs -> ~23 us floor @ 23.3 TB/s. Pure bandwidth.
// Two passes on one stream:
//   1) bulk: branch-free shifted copy via CDNA5 async global<->LDS DMA
//      (clamped src; tail rows get in-bounds garbage on purpose)
//   2) tail: overwrite the 16 appended rows per (b,h) slab from k/v.

typedef float v4f __attribute__((ext_vector_type(4)));
typedef int   v4i __attribute__((ext_vector_type(4)));

#define TPB     256
#define BLOCKS  8192
#define ITEMS   8

#define TPB2    256
#define BLOCKS2 64
#define ITEMS2  4

// Geometry in float4 (16B) units.
static constexpr unsigned SLAB_V4      = 4096u * 128u / 4u;      // 131072 per (b,h) plane
static constexpr unsigned ROW_V4       = 128u / 4u;              // 32
static constexpr unsigned SHIFT_V4     = 16u * ROW_V4;           // 512 (16 new rows)
static constexpr unsigned KEEP_V4      = (4096u - 16u) * ROW_V4; // 130560
static constexpr unsigned TENSOR_V4    = 64u * SLAB_V4;          // 8,388,608 per tensor
static constexpr unsigned GRID_THREADS = (unsigned)TPB * BLOCKS; // 2,097,152
static constexpr unsigned TAIL_THREADS = (unsigned)TPB2 * BLOCKS2; // 16,384
static constexpr unsigned TAIL_V4      = 64u * SHIFT_V4;         // 32,768 per tensor

static_assert((unsigned)(ITEMS / 2) * GRID_THREADS == TENSOR_V4, "bulk tiling exact");
static_assert((unsigned)(ITEMS2 / 2) * TAIL_THREADS == TAIL_V4,  "tail tiling exact");

#if defined(__HIP_DEVICE_COMPILE__) &&                                   \
    __has_builtin(__builtin_amdgcn_global_load_async_to_lds_b128) &&     \
    __has_builtin(__builtin_amdgcn_global_store_async_from_lds_b128)
#define KV_ASYNC 1
#else
#define KV_ASYNC 0
#endif

// Pointer types matching the async builtin prototype:
//   param0: v4i addrspace(1)* (global), param1: v4i addrspace(3)* (LDS)
typedef __attribute__((address_space(1))) v4i as1_v4i;
typedef __attribute__((address_space(3))) v4i as3_v4i;

__device__ __forceinline__ as1_v4i* to_global(const void* p) {
  return (as1_v4i*)(uintptr_t)p;              // flat -> global, same 64-bit value
}
__device__ __forceinline__ as3_v4i* to_lds(const void* p) {
  return (as3_v4i*)(uint32_t)(uintptr_t)p;    // flat LDS addr: low 32 bits = LDS offset
}

__device__ __forceinline__ void wait_async0() {
#if __has_builtin(__builtin_amdgcn_s_wait_asynccnt)
  __builtin_amdgcn_s_wait_asynccnt(0);
#else
  asm volatile("s_wait_asynccnt 0" ::: "memory");
#endif
}

template <int N>
__device__ __forceinline__ void async_copy(const v4f* const (&src)[N],
                                           v4f* const (&dst)[N], v4f* lbuf) {
#if KV_ASYNC
#pragma unroll
  for (int j = 0; j < N; ++j)
    __builtin_amdgcn_global_load_async_to_lds_b128(
        to_global(src[j]), to_lds(lbuf + threadIdx.x + (unsigned)j * blockDim.x), 0, 0);
  wait_async0();
#pragma unroll
  for (int j = 0; j < N; ++j)
    __builtin_amdgcn_global_store_async_from_lds_b128(
        to_global(dst[j]), to_lds(lbuf + threadIdx.x + (unsigned)j * blockDim.x), 0, 0);
  wait_async0();
#else
#pragma unroll
  for (int j = 0; j < N; ++j) {
    v4f d = __builtin_nontemporal_load(src[j]);
    __builtin_nontemporal_store(d, dst[j]);
  }
#endif
}

// Pass 1: branch-free shifted bulk copy. src = base + (t+512), clamped in-bounds.
// The 16 tail rows of every slab receive garbage; pass 2 overwrites them.
__global__ __launch_bounds__(TPB) void kv_bulk_kernel(
    const v4f* __restrict__ ck, const v4f* __restrict__ cv,
    v4f* __restrict__ ok, v4f* __restrict__ ov)
{
  const unsigned gtid = blockIdx.x * (unsigned)TPB + threadIdx.x;
  const v4f* src[ITEMS];
  v4f*       dst[ITEMS];
#pragma unroll
  for (int j = 0; j < ITEMS; ++j) {
    const bool isK = (j < ITEMS / 2);                 // compile-time under unroll
    const v4f* ibase = isK ? ck : cv;
    v4f*       obase = isK ? ok : ov;
    const unsigned t = gtid + (unsigned)(isK ? j : j - ITEMS / 2) * GRID_THREADS;
    unsigned s = t + SHIFT_V4;
    s = (s < TENSOR_V4) ? s : (TENSOR_V4 - 1u);       // v_min_u32: keep read in-bounds
    src[j] = (const v4f*)((const char*)ibase + s * 16u);
    dst[j] = (v4f*)((char*)obase + t * 16u);
  }
  __shared__ v4f lbuf[TPB * ITEMS];                   // 32 KB
  async_copy(src, dst, lbuf);
}

// Pass 2: write the appended 16 rows per slab from k/v into rows [4080,4096).
__global__ __launch_bounds__(TPB2) void kv_tail_kernel(
    const v4f* __restrict__ nk, const v4f* __restrict__ nv,
    v4f* __restrict__ ok, v4f* __restrict__ ov)
{
  const unsigned gtid = blockIdx.x * (unsigned)TPB2 + threadIdx.x;  // [0, 16384)
  const v4f* src[ITEMS2];
  v4f*       dst[ITEMS2];
#pragma unroll
  for (int j = 0; j < ITEMS2; ++j) {
    const bool isK = (j < ITEMS2 / 2);
    const v4f* tbase = isK ? nk : nv;
    v4f*       obase = isK ? ok : ov;
    const unsigned w = gtid + (unsigned)(isK ? j : j - ITEMS2 / 2) * TAIL_THREADS; // [0,32768)
    const unsigned slab = w >> 9;                     // w / SHIFT_V4
    const unsigned rr   = w & (SHIFT_V4 - 1u);
    src[j] = (const v4f*)((const char*)tbase + w * 16u);   // k/v are slab-major contiguous
    dst[j] = (v4f*)((char*)obase + (slab * SLAB_V4 + KEEP_V4 + rr) * 16u);
  }
  __shared__ v4f lbuf[TPB2 * ITEMS2];                 // 16 KB
  async_copy(src, dst, lbuf);
}

extern "C" void kernel_launch(void* const* d_in, const int* in_sizes, int n_in,
                              void* d_out, int out_size, void* d_ws, size_t ws_size,
                              hipStream_t stream) {
  (void)in_sizes; (void)n_in; (void)out_size; (void)d_ws; (void)ws_size;
  const v4f* ck = (const v4f*)d_in[0];
  const v4f* cv = (const v4f*)d_in[1];
  const v4f* nk = (const v4f*)d_in[2];
  const v4f* nv = (const v4f*)d_in[3];
  float* out = (float*)d_out;
  v4f* ok = (v4f*)out;
  v4f* ov = (v4f*)(out + (size_t)TENSOR_V4 * 4u);  // new_k then new_v, flat
  hipLaunchKernelGGL(kv_bulk_kernel, dim3(BLOCKS), dim3(TPB), 0, stream,
                     ck, cv, ok, ov);
  hipLaunchKernelGGL(kv_tail_kernel, dim3(BLOCKS2), dim3(TPB2), 0, stream,
                     nk, nv, ok, ov);                // stream-ordered after bulk
}